// MRLLoss_54159537602915
// MI455X (gfx1250) — compile-verified
//
#include <hip/hip_runtime.h>
#include <math.h>

typedef __attribute__((ext_vector_type(2))) float v2f;
typedef __attribute__((ext_vector_type(8))) float v8f;

#define NROWS 8192
#define KCOLS 30
#define CHUNK 2048   // j-tile staged in LDS: 2 arrays * 8KB = 16KB
#define P2_BLOCKS (NROWS / 128)

// ---------------- Phase 1: per-row MRL + extract t / e ----------------
__global__ __launch_bounds__(256) void mrl_phase1(
    const float* __restrict__ est, const float* __restrict__ target,
    const float* __restrict__ intervals,
    float* __restrict__ m, float* __restrict__ t, float* __restrict__ e) {
  int i = blockIdx.x * 256 + threadIdx.x;
  if (i >= NROWS) return;
  const float* row = est + (long)i * KCOLS;
  float sp = 1.0f, acc = 0.0f;
#pragma unroll
  for (int k = 0; k < KCOLS; ++k) {
    float h = 1.0f / (1.0f + __expf(-row[k]));   // sigmoid
    acc += intervals[k] * (h * sp);              // intervals * pdf
    sp *= (1.0f - h);                            // running cumprod(1-h)
  }
  m[i] = acc;
  t[i] = target[2 * i];
  e[i] = (target[2 * i + 1] != 0.0f) ? 1.0f : 0.0f;
}

// ---------------- Phase 2: pairwise sums via f32 WMMA row-sum trick ----------------
// Wave w of block b owns rows i0..i0+15. A[i,kk] = e_i * (t_i < t_j), B = ones:
//   accG[row] += #comparable j  (replicated across 16 column lanes)
//   accH[row] += sum of mrl_j over comparable j
__global__ __launch_bounds__(256) void mrl_phase2(
    const float* __restrict__ m, const float* __restrict__ t,
    const float* __restrict__ e,
    float* __restrict__ sPartOut, float* __restrict__ cPartOut) {
  __shared__ float tC[CHUNK];
  __shared__ float mC[CHUNK];
  __shared__ float wS[8];
  __shared__ float wC[8];

  const int tid  = threadIdx.x;
  const int lane = tid & 31;
  const int wave = tid >> 5;
  const int i0   = blockIdx.x * 128 + wave * 16;

  // A-matrix 16x4 f32 layout: lane&15 = M row; lanes 0-15 hold K=0,1; lanes 16-31 hold K=2,3
  const int row  = lane & 15;
  const int kk0  = (lane < 16) ? 0 : 2;
  const float ti = t[i0 + row];
  const float em = e[i0 + row];   // 0.0 or 1.0, folded into the mask

  v8f accG = {};
  v8f accH = {};
  v2f onesB;
  onesB.x = 1.0f; onesB.y = 1.0f; // all-ones B: layout-independent

  for (int base = 0; base < NROWS; base += CHUNK) {
    __syncthreads();
    for (int s = tid; s < CHUNK; s += 256) {
      tC[s] = t[base + s];
      mC[s] = m[base + s];
    }
    __syncthreads();
    for (int jj = 0; jj < CHUNK; jj += 4) {
      const int j = jj + kk0;                 // even -> b64-friendly LDS access
      const float tj0 = tC[j], tj1 = tC[j + 1];
      const float mj0 = mC[j], mj1 = mC[j + 1];
      v2f a, aw;
      a.x  = (ti < tj0) ? em : 0.0f;          // comparable mask (fp32-exact 0/1)
      a.y  = (ti < tj1) ? em : 0.0f;
      aw.x = a.x * mj0;                       // mask * mrl_j
      aw.y = a.y * mj1;
      accG = __builtin_amdgcn_wmma_f32_16x16x4_f32(
          false, a,  false, onesB, (short)0, accG, false, false);
      accH = __builtin_amdgcn_wmma_f32_16x16x4_f32(
          false, aw, false, onesB, (short)0, accH, false, false);
    }
  }

  // D layout: VGPR v holds row (v | (lane>=16 ? 8 : 0)); value replicated over 16 column lanes
  const int rbase = (lane < 16) ? 0 : 8;
  float sP = 0.0f, cP = 0.0f;
#pragma unroll
  for (int v = 0; v < 8; ++v) {
    const float mi = m[i0 + rbase + v];
    sP += mi * accG[v] - accH[v];   // e_i*(m_i*G_i - H_i), 16x replicated
    cP += accG[v];                  // e_i*G_i,            16x replicated
  }
  // wave reduction (16x replication cancels in the final S/cnt ratio)
  for (int off = 16; off >= 1; off >>= 1) {
    sP += __shfl_xor(sP, off, 32);
    cP += __shfl_xor(cP, off, 32);
  }
  if (lane == 0) { wS[wave] = sP; wC[wave] = cP; }
  __syncthreads();
  if (tid == 0) {
    float S = 0.0f, C = 0.0f;
#pragma unroll
    for (int w = 0; w < 8; ++w) { S += wS[w]; C += wC[w]; }
    sPartOut[blockIdx.x] = S;
    cPartOut[blockIdx.x] = C;
  }
}

// ---------------- Phase 3: deterministic final reduce + divide ----------------
__global__ __launch_bounds__(64) void mrl_phase3(
    const float* __restrict__ sPart, const float* __restrict__ cPart,
    float* __restrict__ out) {
  if (threadIdx.x == 0) {
    float S = 0.0f, C = 0.0f;
    for (int b = 0; b < P2_BLOCKS; ++b) { S += sPart[b]; C += cPart[b]; }
    out[0] = S / C;
  }
}

extern "C" void kernel_launch(void* const* d_in, const int* in_sizes, int n_in,
                              void* d_out, int out_size, void* d_ws, size_t ws_size,
                              hipStream_t stream) {
  (void)in_sizes; (void)n_in; (void)out_size; (void)ws_size;
  const float* est       = (const float*)d_in[0];   // (8192, 30) f32
  const float* target    = (const float*)d_in[1];   // (8192, 2)  f32
  const float* intervals = (const float*)d_in[2];   // (30,)      f32
  float* out = (float*)d_out;

  // workspace layout (floats): m[N] | t[N] | e[N] | sPart[64] | cPart[64]
  float* ws    = (float*)d_ws;
  float* wm    = ws;
  float* wt    = ws + NROWS;
  float* we    = ws + 2 * NROWS;
  float* sPart = ws + 3 * NROWS;
  float* cPart = ws + 3 * NROWS + P2_BLOCKS;

  mrl_phase1<<<NROWS / 256, 256, 0, stream>>>(est, target, intervals, wm, wt, we);
  mrl_phase2<<<P2_BLOCKS, 256, 0, stream>>>(wm, wt, we, sPart, cPart);
  mrl_phase3<<<1, 64, 0, stream>>>(sPart, cPart, out);
}